// ONNXScatter_32487132627633
// MI455X (gfx1250) — compile-verified
//
#include <hip/hip_runtime.h>
#include <hip/hip_bf16.h>

// segment_sum: src (1250000, 64) f32, index (1250000,) int, out (50000, 64) f32.
// HBM-bound: 320 MB src stream @ 23.3 TB/s => ~14us floor. Output (12.8 MB)
// lives in 192 MB L2; atomics resolve there. Strategy:
//   - b128 (float4) NT loads of src (streamed once, keep L2 for output)
//   - non-returning device-scope global_atomic_add_f32 into out
//   - zero-init out first (harness poisons it)

typedef float f32x4 __attribute__((ext_vector_type(4)));

__global__ __launch_bounds__(256) void zero_f32x4_kernel(f32x4* __restrict__ out,
                                                         int nquads) {
    int i = blockIdx.x * blockDim.x + threadIdx.x;
    if (i < nquads) {
        f32x4 z = {0.0f, 0.0f, 0.0f, 0.0f};
        __builtin_nontemporal_store(z, out + i);   // global_store_b128, NT
    }
}

__global__ __launch_bounds__(256) void segment_sum_kernel(
    const float* __restrict__ src,
    const int*   __restrict__ index,
    float*       __restrict__ out,
    int nrows) {
    // One float4 per lane; 16 lanes cover one 64-float row.
    int qid = blockIdx.x * blockDim.x + threadIdx.x;
    int row = qid >> 4;
    if (row >= nrows) return;
    int col = (qid & 15) << 2;

    // Streaming loads: non-temporal so the 320 MB src stream does not evict
    // the 12.8 MB output working set from L2.
    const f32x4* sp = (const f32x4*)(src + (size_t)row * 64 + col);
    f32x4 v = __builtin_nontemporal_load(sp);          // global_load_b128 th:NT
    int idx  = __builtin_nontemporal_load(index + row); // merged across 16 lanes

    float* op = out + (size_t)idx * 64 + col;
    // Non-returning f32 atomics (relaxed, agent scope) -> global_atomic_add_f32,
    // tracked on STOREcnt, resolved at the L2 atomic units.
    __hip_atomic_fetch_add(op + 0, v.x, __ATOMIC_RELAXED, __HIP_MEMORY_SCOPE_AGENT);
    __hip_atomic_fetch_add(op + 1, v.y, __ATOMIC_RELAXED, __HIP_MEMORY_SCOPE_AGENT);
    __hip_atomic_fetch_add(op + 2, v.z, __ATOMIC_RELAXED, __HIP_MEMORY_SCOPE_AGENT);
    __hip_atomic_fetch_add(op + 3, v.w, __ATOMIC_RELAXED, __HIP_MEMORY_SCOPE_AGENT);
}

extern "C" void kernel_launch(void* const* d_in, const int* in_sizes, int n_in,
                              void* d_out, int out_size, void* d_ws, size_t ws_size,
                              hipStream_t stream) {
    const float* src   = (const float*)d_in[0];
    const int*   index = (const int*)d_in[1];
    // d_in[2] = dim_size scalar; out_size already encodes dim_size * 64.
    float* out = (float*)d_out;

    const int row_width = 64;
    const int nrows = in_sizes[0] / row_width;   // 1,250,000

    // 1) Zero the output (harness poisons d_out with 0xAA).
    int out_quads = out_size / 4;                // 800,000 float4s
    int zb = (out_quads + 255) / 256;
    zero_f32x4_kernel<<<zb, 256, 0, stream>>>((f32x4*)out, out_quads);

    // 2) Scatter-add: one float4 per thread, 16 threads per row.
    long long quads = (long long)nrows * 16;     // 20,000,000
    int sb = (int)((quads + 255) / 256);
    segment_sum_kernel<<<sb, 256, 0, stream>>>(src, index, out, nrows);
}